// MinibatchDiscrimination1d_42949672961129
// MI455X (gfx1250) — compile-verified
//
#include <hip/hip_runtime.h>
#include <hip/hip_bf16.h>

// Problem constants from the reference
#define NN 512   // batch
#define AA 512   // input features / GEMM K
#define BB 32    // kernels
#define CC 16    // dims per kernel
#define OUTW (AA + BB)  // 544, output row stride

typedef __attribute__((ext_vector_type(2))) float v2f;
typedef __attribute__((ext_vector_type(8))) float v8f;

// ---------------------------------------------------------------------------
// Kernel 1: M = x @ W  (W = T viewed as (A, B*C) row-major), full fp32 via
// V_WMMA_F32_16X16X4_F32. One wave per 16x16 output tile; K swept 4 at a time.
//
// Fragment layouts per CDNA5 ISA 7.12.2 (wave32):
//   A (16x4 f32, 2 VGPRs): lane L holds row m=L&15; VGPR0=K=kb, VGPR1=K=kb+1
//                          where kb = (L>>4)*2.
//   B (4x16 f32, 2 VGPRs): lane L holds col n=L&15; VGPR0=K=kb, VGPR1=K=kb+1.
//   C/D (16x16 f32, 8 VGPRs): VGPR r holds row r + (L>>4)*8, col n=L&15.
// ---------------------------------------------------------------------------
__global__ __launch_bounds__(32) void mbd_gemm_wmma_f32(
    const float* __restrict__ X,   // (N, A) row-major
    const float* __restrict__ W,   // (A, B*C) row-major
    float* __restrict__ M)         // (N, B*C) row-major
{
    const int tile_n = blockIdx.x;         // 0..31 : column tile of M
    const int tile_m = blockIdx.y;         // 0..31 : row tile of M
    const int lane   = threadIdx.x;        // 0..31
    const int half   = lane >> 4;          // 0 or 1
    const int l16    = lane & 15;
    const int kb     = half * 2;           // K sub-offset for this lane half

    const float* xrow = X + (tile_m * 16 + l16) * (size_t)AA;  // A row
    const float* wcol = W + tile_n * 16 + l16;                  // B column base

    v8f acc = {};
    #pragma unroll 4
    for (int k0 = 0; k0 < AA; k0 += 4) {
        v2f a, b;
        a.x = xrow[k0 + kb];
        a.y = xrow[k0 + kb + 1];
        b.x = wcol[(size_t)(k0 + kb) * (BB * CC)];
        b.y = wcol[(size_t)(k0 + kb + 1) * (BB * CC)];
        // (neg_a, A, neg_b, B, c_mod, C, reuse_a, reuse_b)
        acc = __builtin_amdgcn_wmma_f32_16x16x4_f32(
            false, a, false, b, (short)0, acc, false, false);
    }

    float* mout = M + (size_t)(tile_m * 16 + half * 8) * (BB * CC)
                    + tile_n * 16 + l16;
    #pragma unroll
    for (int r = 0; r < 8; ++r)
        mout[(size_t)r * (BB * CC)] = acc[r];
}

// ---------------------------------------------------------------------------
// Kernel 2: out[i, A+b] = sum_j exp(-sum_c |M[i,b,c]-M[j,b,c]|) - 1
// Grid (BB, N/128), 128 threads. Stage the 512x16 slice M[:,b,:] (32 KB)
// into LDS via CDNA5 async global->LDS (no VGPR round-trip, ASYNCcnt-tracked),
// then sweep j with wave-uniform LDS broadcast reads.
// ---------------------------------------------------------------------------
__global__ __launch_bounds__(128) void mbd_pairwise_l1exp(
    const float* __restrict__ M,   // (N, B*C)
    float* __restrict__ out)       // (N, A+B)
{
    const int b   = blockIdx.x;    // 0..BB-1
    const int ic  = blockIdx.y;    // 0..3 (i-chunk)
    const int tid = threadIdx.x;   // 0..127

    __shared__ float4 sM[NN * (CC / 4)];   // 512 rows x 16 floats = 32 KB

    // Async stage of M[:, b, 0:16]: 2048 16B packets, 16 per lane.
    // global_load_async_to_lds_b128: VDST = LDS byte offset (wave-relative,
    // = low 32 bits of the generic shared address), VADDR = 64-bit global addr.
    for (int e = tid; e < NN * 4; e += 128) {
        const int row = e >> 2;            // 0..511
        const int q   = e & 3;             // float4 within row
        const float4* gsrc = (const float4*)(M + (size_t)row * (BB * CC) + b * CC) + q;
        unsigned loff = (unsigned)(uintptr_t)(&sM[e]);
        asm volatile("global_load_async_to_lds_b128 %0, %1, off"
                     :: "v"(loff), "v"(gsrc)
                     : "memory");
    }
    // Drain this wave's async-to-LDS transfers, then barrier for whole WG.
    asm volatile("s_wait_asynccnt 0x0" ::: "memory");
    __syncthreads();

    const int i = ic * 128 + tid;
    const float4 m0 = sM[i * 4 + 0];
    const float4 m1 = sM[i * 4 + 1];
    const float4 m2 = sM[i * 4 + 2];
    const float4 m3 = sM[i * 4 + 3];

    float acc = 0.0f;
    #pragma unroll 4
    for (int j = 0; j < NN; ++j) {
        const float4 n0 = sM[j * 4 + 0];
        const float4 n1 = sM[j * 4 + 1];
        const float4 n2 = sM[j * 4 + 2];
        const float4 n3 = sM[j * 4 + 3];
        float d = fabsf(m0.x - n0.x) + fabsf(m0.y - n0.y)
                + fabsf(m0.z - n0.z) + fabsf(m0.w - n0.w)
                + fabsf(m1.x - n1.x) + fabsf(m1.y - n1.y)
                + fabsf(m1.z - n1.z) + fabsf(m1.w - n1.w)
                + fabsf(m2.x - n2.x) + fabsf(m2.y - n2.y)
                + fabsf(m2.z - n2.z) + fabsf(m2.w - n2.w)
                + fabsf(m3.x - n3.x) + fabsf(m3.y - n3.y)
                + fabsf(m3.z - n3.z) + fabsf(m3.w - n3.w);
        acc += __expf(-d);   // v_exp_f32 transcendental
    }
    out[(size_t)i * OUTW + AA + b] = acc - 1.0f;   // subtract self term exp(0)
}

// ---------------------------------------------------------------------------
// Kernel 3: out[i, 0:512] = x[i, :]  (vectorized float4 copy, stride 544)
// ---------------------------------------------------------------------------
__global__ __launch_bounds__(256) void mbd_copy_x(
    const float* __restrict__ X,
    float* __restrict__ out)
{
    const int idx = blockIdx.x * 256 + threadIdx.x;  // over N * (A/4) float4s
    const int i   = idx >> 7;        // row (128 float4 per row)
    const int c4  = idx & 127;       // float4 column
    ((float4*)(out + (size_t)i * OUTW))[c4] =
        ((const float4*)(X + (size_t)i * AA))[c4];
}

extern "C" void kernel_launch(void* const* d_in, const int* in_sizes, int n_in,
                              void* d_out, int out_size, void* d_ws, size_t ws_size,
                              hipStream_t stream) {
    const float* x = (const float*)d_in[0];   // (512, 512) fp32
    const float* T = (const float*)d_in[1];   // (512, 32, 16) fp32 == (512, 512)
    float* out = (float*)d_out;               // (512, 544) fp32
    float* M   = (float*)d_ws;                // scratch: (512, 512) fp32 = 1 MB

    // 1) M = x @ T  (fp32 WMMA, one wave per 16x16 tile)
    dim3 g1(BB * CC / 16, NN / 16);           // 32 x 32 tiles
    mbd_gemm_wmma_f32<<<g1, 32, 0, stream>>>(x, T, M);

    // 2) pairwise exp(-L1) reduction into out[:, 512:544]
    dim3 g2(BB, NN / 128);                    // 32 x 4 blocks
    mbd_pairwise_l1exp<<<g2, 128, 0, stream>>>(M, out);

    // 3) copy x into out[:, 0:512]
    mbd_copy_x<<<(NN * AA / 4) / 256, 256, 0, stream>>>(x, out);
}